// PretrainModel_48155173322953
// MI455X (gfx1250) — compile-verified
//
#include <hip/hip_runtime.h>
#include <hip/hip_bf16.h>

typedef __attribute__((ext_vector_type(2))) float v2f;
typedef __attribute__((ext_vector_type(8))) float v8f;

#define D_FEAT 128
#define NEG_SLOPE 0.01f

// ---------------- degree kernels ----------------
__global__ void init_deg_kernel(float* __restrict__ deg, int n) {
    int i = blockIdx.x * blockDim.x + threadIdx.x;
    if (i < n) deg[i] = 1.0f;  // +1 for implicit self-loop
}

__global__ void count_deg_kernel(const long long* __restrict__ ei, float* __restrict__ deg,
                                 int ecount) {
    int e = blockIdx.x * blockDim.x + threadIdx.x;
    if (e < ecount) {
        long long dst = ei[(size_t)ecount + e];
        atomicAdd(&deg[dst], 1.0f);
    }
}

__global__ void finish_deg_kernel(const float* __restrict__ deg, float* __restrict__ dis,
                                  float* __restrict__ dinv, int n) {
    int i = blockIdx.x * blockDim.x + threadIdx.x;
    if (i < n) {
        float d = deg[i];
        dis[i] = rsqrtf(d);
        dinv[i] = 1.0f / d;
    }
}

// ---------------- WMMA GEMM: Hout = Hin[N,128] @ W[128,128] ----------------
// One wave computes one 16x16 output tile using v_wmma_f32_16x16x4_f32.
// Block = 256 threads (8 waves): 16 rows x 128 cols per block.
// MODE 0: GCN epilogue -> HW = h@W ; ACC = HW*deg_inv[row] + bias[col]
// MODE 1: final FC    -> ACC = relu(h@W + bias)
template <int MODE>
__global__ void gemm_wmma_kernel(const float* __restrict__ Hin, const float* __restrict__ W,
                                 const float* __restrict__ bias, const float* __restrict__ dinv,
                                 float* __restrict__ HW, float* __restrict__ ACC, int nrows) {
    extern __shared__ float lds[];
    float* sW = lds;                 // 128*128 floats
    float* sA = lds + D_FEAT * D_FEAT;  // 16*132 floats (padded rows)

    const int tid = threadIdx.x;
    const int wave = tid >> 5;
    const int lane = tid & 31;
    const int rowBase = blockIdx.x * 16;
    const int colBase = wave * 16;

    // Stage W into LDS (16384 floats) via float4
    {
        const float4* Wv = (const float4*)W;
        float4* sWv = (float4*)sW;
        for (int i = tid; i < D_FEAT * D_FEAT / 4; i += 256) sWv[i] = Wv[i];
    }
    // Stage 16-row A tile into LDS (row pitch 132 to spread banks)
    {
        for (int i = tid; i < 16 * 32; i += 256) {
            int r = i >> 5;       // 0..15
            int c4 = i & 31;      // float4 column
            int row = rowBase + r;
            float4 v = make_float4(0.f, 0.f, 0.f, 0.f);
            if (row < nrows) v = ((const float4*)(Hin + (size_t)row * D_FEAT))[c4];
            float* p = &sA[r * 132 + c4 * 4];
            p[0] = v.x; p[1] = v.y; p[2] = v.z; p[3] = v.w;
        }
    }
    __syncthreads();

    // A 16x4 layout: lanes 0-15 = M, VGPR0/1 = K0/K1 ; lanes 16-31 = K2/K3
    // B 4x16 layout: lanes 0-15 = N, VGPR0/1 = K0/K1 ; lanes 16-31 = K2/K3
    const int half = lane >> 4;
    const int l16 = lane & 15;

    v8f c = {};
#pragma unroll
    for (int k = 0; k < D_FEAT; k += 4) {
        const int kk = k + 2 * half;
        v2f a, b;
        a.x = sA[l16 * 132 + kk];
        a.y = sA[l16 * 132 + kk + 1];
        b.x = sW[kk * D_FEAT + colBase + l16];
        b.y = sW[(kk + 1) * D_FEAT + colBase + l16];
        c = __builtin_amdgcn_wmma_f32_16x16x4_f32(false, a, false, b, (short)0, c, false, false);
    }

    // C/D layout: VGPR r, lanes 0-15 -> M=r, N=lane ; lanes 16-31 -> M=r+8
    const int col = colBase + l16;
    const float bc = bias[col];
#pragma unroll
    for (int r = 0; r < 8; ++r) {
        const int row = rowBase + r + 8 * half;
        if (row < nrows) {
            const float v = c[r];
            const size_t idx = (size_t)row * D_FEAT + col;
            if (MODE == 0) {
                HW[idx] = v;
                ACC[idx] = v * dinv[row] + bc;  // self-loop term + bias
            } else {
                const float o = v + bc;
                ACC[idx] = o > 0.f ? o : 0.f;   // relu
            }
        }
    }
}

// ---------------- edge scatter: one wave per edge ----------------
// ACC[dst] += HW[src] * (dis[src]*dis[dst]); lane handles 4 floats.
__global__ void edge_scatter_kernel(const float* __restrict__ HW,
                                    const long long* __restrict__ ei,
                                    const float* __restrict__ dis,
                                    float* __restrict__ ACC, int ecount) {
    const int gwave = (int)((blockIdx.x * (size_t)blockDim.x + threadIdx.x) >> 5);
    const int lane = threadIdx.x & 31;
    if (gwave >= ecount) return;
    const long long s = ei[gwave];
    const long long d = ei[(size_t)ecount + gwave];
    const float norm = dis[s] * dis[d];
    const float4 v = ((const float4*)(HW + (size_t)s * D_FEAT))[lane];
    float* out = ACC + (size_t)d * D_FEAT + lane * 4;
    atomicAdd(out + 0, v.x * norm);
    atomicAdd(out + 1, v.y * norm);
    atomicAdd(out + 2, v.z * norm);
    atomicAdd(out + 3, v.w * norm);
}

// ---------------- leaky relu (float4 vectorized) ----------------
__global__ void leaky_kernel(float* __restrict__ a, int nvec4) {
    int i = blockIdx.x * blockDim.x + threadIdx.x;
    if (i < nvec4) {
        float4 v = ((float4*)a)[i];
        v.x = v.x >= 0.f ? v.x : NEG_SLOPE * v.x;
        v.y = v.y >= 0.f ? v.y : NEG_SLOPE * v.y;
        v.z = v.z >= 0.f ? v.z : NEG_SLOPE * v.z;
        v.w = v.w >= 0.f ? v.w : NEG_SLOPE * v.w;
        ((float4*)a)[i] = v;
    }
}

extern "C" void kernel_launch(void* const* d_in, const int* in_sizes, int n_in,
                              void* d_out, int out_size, void* d_ws, size_t ws_size,
                              hipStream_t stream) {
    const float* x = (const float*)d_in[0];
    const long long* ei = (const long long*)d_in[1];
    const float* Wt[6];
    const float* bt[6];
    for (int i = 0; i < 5; ++i) {
        Wt[i] = (const float*)d_in[2 + 2 * i];
        bt[i] = (const float*)d_in[3 + 2 * i];
    }
    Wt[5] = (const float*)d_in[12];
    bt[5] = (const float*)d_in[13];

    const int N = in_sizes[0] / D_FEAT;
    const int E = in_sizes[1] / 2;

    float* out = (float*)d_out;
    float* gene = out;                        // [N,128] gene_predictions
    float* hidden = out + (size_t)N * D_FEAT; // [N,128] hidden_state

    float* ws = (float*)d_ws;
    float* deg = ws;
    float* dis = deg + N;
    float* dinv = dis + N;
    float* HW = dinv + N;                     // [N,128] gemm output
    float* F0 = HW + (size_t)N * D_FEAT;      // ping
    float* F1 = F0 + (size_t)N * D_FEAT;      // pong

    // ---- degrees (with self-loops) ----
    init_deg_kernel<<<(N + 255) / 256, 256, 0, stream>>>(deg, N);
    count_deg_kernel<<<(E + 255) / 256, 256, 0, stream>>>(ei, deg, E);
    finish_deg_kernel<<<(N + 255) / 256, 256, 0, stream>>>(deg, dis, dinv, N);

    const size_t ldsBytes = (size_t)(D_FEAT * D_FEAT + 16 * 132) * sizeof(float);
    const int gemmBlocks = (N + 15) / 16;
    const int scatterBlocks = (E + 7) / 8;  // 8 waves (edges) per 256-thread block
    const int nvec4 = N * (D_FEAT / 4);

    const float* hin = x;
    for (int layer = 0; layer < 5; ++layer) {
        float* acc = (layer == 4) ? hidden : ((layer & 1) ? F1 : F0);
        gemm_wmma_kernel<0><<<gemmBlocks, 256, ldsBytes, stream>>>(hin, Wt[layer], bt[layer],
                                                                   dinv, HW, acc, N);
        edge_scatter_kernel<<<scatterBlocks, 256, 0, stream>>>(HW, ei, dis, acc, E);
        if (layer < 4)
            leaky_kernel<<<(nvec4 + 255) / 256, 256, 0, stream>>>(acc, nvec4);
        hin = acc;
    }

    // gene_predictions = relu(hidden @ Wg + bg)
    gemm_wmma_kernel<1><<<gemmBlocks, 256, ldsBytes, stream>>>(hidden, Wt[5], bt[5], dinv,
                                                               nullptr, gene, N);
}